// GraphConvolutionLayer_28114855919650
// MI455X (gfx1250) — compile-verified
//
#include <hip/hip_runtime.h>
#include <math.h>

// Problem constants (from reference)
#define BB   2
#define NN   512
#define DD   128
#define BN   (BB*NN)          // 1024
#define EPSF 1e-5f
#define WP   132              // padded LDS row pitch (floats) -> conflict-free stride

typedef __attribute__((ext_vector_type(2))) float v2f;
typedef __attribute__((ext_vector_type(8))) float v8f;

// One 16x16 f32 output tile via 32 x V_WMMA_F32_16X16X4_F32.
// A: 16 rows x 128 cols in LDS (pitch WP). Bw: weight rows (out-dim major) in LDS (pitch WP),
// so Bfrag[K][Ncol] = Bw[nn*WP + K].
static __device__ __forceinline__ v8f tile_mm128(const float* __restrict__ A,
                                                 const float* __restrict__ Bw,
                                                 int lr, int kk, int nn) {
  v8f acc = {0.f,0.f,0.f,0.f,0.f,0.f,0.f,0.f};
#pragma unroll
  for (int k = 0; k < 128; k += 4) {
    v2f a = *(const v2f*)&A[lr * WP + k + kk];
    v2f b = *(const v2f*)&Bw[nn * WP + k + kk];
    acc = __builtin_amdgcn_wmma_f32_16x16x4_f32(false, a, false, b, (short)0, acc,
                                                false, false);
  }
  return acc;
}

// ---------------------------------------------------------------------------
// K1 / K1b: OUT[1024,128] = IN[1024,128] @ W[:, :128]^T + bias   (row pitch wpitch)
// grid 64 blocks (16 rows each), 256 threads = 8 waves = 8 N-tiles.
// ---------------------------------------------------------------------------
__global__ void gemm_wmma(float* __restrict__ out, const float* __restrict__ in,
                          const float* __restrict__ W, int wpitch,
                          const float* __restrict__ bias) {
  extern __shared__ float smem[];
  float* Wl = smem;               // 128*WP
  float* At = smem + 128 * WP;    // 16*WP
  const int t = threadIdx.x;
  const int r0 = blockIdx.x * 16;

  for (int idx = t; idx < 128 * 128; idx += 256) {
    int rr = idx >> 7, cc = idx & 127;
    Wl[rr * WP + cc] = W[rr * wpitch + cc];
  }
  for (int idx = t; idx < 16 * 128; idx += 256) {
    int rr = idx >> 7, cc = idx & 127;
    At[rr * WP + cc] = in[(size_t)(r0 + rr) * 128 + cc];
  }
  __syncthreads();

  const int lane = t & 31, w = t >> 5;
  const int n0 = w * 16, lr = lane & 15, hi = lane >> 4, kk = hi * 2;
  const int nn = n0 + lr;

  v8f acc = tile_mm128(At, Wl, lr, kk, nn);

  const float bv = bias[nn];
#pragma unroll
  for (int r = 0; r < 8; ++r) {
    int m = r + 8 * hi;
    out[(size_t)(r0 + m) * 128 + nn] = acc[r] + bv;
  }
}

// ---------------------------------------------------------------------------
// K1c: per-node attention dots. u[i] = <W_att[0:128], h_i>, v[i] = <W_att[128:256], h_i>
// grid 1024 blocks, 128 threads.
// ---------------------------------------------------------------------------
__global__ void uv_kernel(const float* __restrict__ h, const float* __restrict__ Watt,
                          float* __restrict__ u, float* __restrict__ v) {
  __shared__ float su[128], sv[128];
  const int i = blockIdx.x, t = threadIdx.x;
  float hv = h[(size_t)i * 128 + t];
  su[t] = hv * Watt[t];
  sv[t] = hv * Watt[128 + t];
  __syncthreads();
  for (int off = 64; off > 0; off >>= 1) {
    if (t < off) { su[t] += su[t + off]; sv[t] += sv[t + off]; }
    __syncthreads();
  }
  if (t == 0) { u[i] = su[0]; v[i] = sv[0]; }
}

// ---------------------------------------------------------------------------
// K2: edge aggregation (the dominant, non-GEMM part).
//   s[i,:] = sum_j att_ij * relu(p_j + a_ij * wcol),  c[i] = sum_j att_ij
// One block per row i (1024 blocks). 256 threads: halves split the j-range.
// att_j / a_j staged once in LDS; p reads are coalesced and L2-resident.
// ---------------------------------------------------------------------------
__global__ void pair_agg(const float* __restrict__ adj, const float* __restrict__ p,
                         const float* __restrict__ u, const float* __restrict__ v,
                         const float* __restrict__ Wm1, const float* __restrict__ Watt,
                         const float* __restrict__ batt,
                         float* __restrict__ s, float* __restrict__ c) {
  __shared__ float att_s[NN], a_s[NN], sred[128], red[256];
  const int bi = blockIdx.x;            // 0..1023  (= b*N + i)
  const int t = threadIdx.x;            // 0..255
  const int vbase = bi & ~(NN - 1);     // start of this batch's nodes
  const float ui = u[bi];
  const float wa = Watt[256];
  const float ba = batt[0];
  const float* arow = adj + (size_t)bi * NN;

  // stage attention row
  for (int j = t; j < NN; j += 256) {
    float a = arow[j];
    float x = ui + v[vbase + j] + wa * a + ba;
    att_s[j] = 1.0f / (1.0f + expf(-x));
    a_s[j] = a;
  }
  __syncthreads();

  // c = sum_j att_j (tree reduction)
  red[t] = att_s[t] + att_s[t + 256];
  __syncthreads();
  for (int off = 128; off > 0; off >>= 1) {
    if (t < off) red[t] += red[t + off];
    __syncthreads();
  }
  if (t == 0) c[bi] = red[0];

  // main accumulation
  const int d = t & 127, half = t >> 7;
  const float wc = Wm1[d * 129 + 128];          // W_m1[:, D] column
  const float* pb = p + (size_t)vbase * 128 + d;
  float acc = 0.f;
  const int j0 = half * 256;
#pragma unroll 4
  for (int jj = 0; jj < 256; ++jj) {
    int j = j0 + jj;
    __builtin_prefetch(&pb[(size_t)(j + 8) * 128], 0, 0);
    float pv = pb[(size_t)j * 128];
    float tm = fmaxf(fmaf(a_s[j], wc, pv), 0.f);
    acc = fmaf(att_s[j], tm, acc);
  }
  if (half == 1) sred[d] = acc;
  __syncthreads();
  if (half == 0) s[(size_t)bi * 128 + d] = acc + sred[d];
}

// ---------------------------------------------------------------------------
// K3: fused output network + residual + LayerNorm + ReLU.
//   agg = s@Wm2^T + c*b_m2 ; t1 = relu(agg@Wo1^T + b_o1) ; t2 = t1@Wo2^T + b_o2
//   out = relu(LN(h + t2))
// All three 64KB weights live in LDS (~216KB total, within 320KB/WGP).
// grid 64 blocks (16 rows each), 256 threads = 8 waves = 8 N-tiles per stage.
// ---------------------------------------------------------------------------
__global__ void out_net(float* __restrict__ out, const float* __restrict__ s,
                        const float* __restrict__ csum, const float* __restrict__ h,
                        const float* __restrict__ Wm2, const float* __restrict__ bm2,
                        const float* __restrict__ Wo1, const float* __restrict__ bo1,
                        const float* __restrict__ Wo2, const float* __restrict__ bo2,
                        const float* __restrict__ lng, const float* __restrict__ lnb) {
  extern __shared__ float smem[];
  float* W0 = smem;                 // 128*WP
  float* W1 = W0 + 128 * WP;
  float* W2 = W1 + 128 * WP;
  float* A0 = W2 + 128 * WP;        // 16*WP ping
  float* A1 = A0 + 16 * WP;         // 16*WP pong
  float* cr  = A1 + 16 * WP;        // 16
  float* red = cr + 16;             // 16*17
  float* mus = red + 16 * 17;       // 16
  float* rsd = mus + 16;            // 16

  const int t = threadIdx.x;
  const int r0 = blockIdx.x * 16;

  for (int idx = t; idx < 16384; idx += 256) {
    int rr = idx >> 7, cc = idx & 127;
    W0[rr * WP + cc] = Wm2[idx];
    W1[rr * WP + cc] = Wo1[idx];
    W2[rr * WP + cc] = Wo2[idx];
  }
  for (int idx = t; idx < 2048; idx += 256) {
    int rr = idx >> 7, cc = idx & 127;
    A0[rr * WP + cc] = s[(size_t)(r0 + rr) * 128 + cc];
  }
  if (t < 16) cr[t] = csum[r0 + t];
  __syncthreads();

  const int lane = t & 31, w = t >> 5;
  const int n0 = w * 16, lr = lane & 15, hi = lane >> 4, kk = hi * 2;
  const int nn = n0 + lr;

  // stage 1: agg = s @ Wm2^T + c*b_m2
  {
    v8f acc = tile_mm128(A0, W0, lr, kk, nn);
    float bm = bm2[nn];
#pragma unroll
    for (int r = 0; r < 8; ++r) { int m = r + 8 * hi; A1[m * WP + nn] = acc[r] + cr[m] * bm; }
  }
  __syncthreads();
  // stage 2: t1 = relu(agg @ Wo1^T + b_o1)
  {
    v8f acc = tile_mm128(A1, W1, lr, kk, nn);
    float b1 = bo1[nn];
#pragma unroll
    for (int r = 0; r < 8; ++r) { int m = r + 8 * hi; A0[m * WP + nn] = fmaxf(acc[r] + b1, 0.f); }
  }
  __syncthreads();
  // stage 3: u = h + (t1 @ Wo2^T + b_o2)
  {
    v8f acc = tile_mm128(A0, W2, lr, kk, nn);
    float b2 = bo2[nn];
#pragma unroll
    for (int r = 0; r < 8; ++r) {
      int m = r + 8 * hi;
      A1[m * WP + nn] = acc[r] + b2 + h[(size_t)(r0 + m) * 128 + nn];
    }
  }
  __syncthreads();

  // LayerNorm over last dim (16 rows x 128 cols), then ReLU, write out.
  const int row = t >> 4, c16 = t & 15;
  float part = 0.f;
  for (int q = 0; q < 8; ++q) part += A1[row * WP + c16 * 8 + q];
  red[row * 17 + c16] = part;
  __syncthreads();
  if (t < 16) {
    float mu = 0.f;
    for (int c2 = 0; c2 < 16; ++c2) mu += red[t * 17 + c2];
    mus[t] = mu * (1.f / 128.f);
  }
  __syncthreads();
  float mu = mus[row];
  part = 0.f;
  for (int q = 0; q < 8; ++q) { float x = A1[row * WP + c16 * 8 + q] - mu; part += x * x; }
  red[row * 17 + c16] = part;
  __syncthreads();
  if (t < 16) {
    float vv = 0.f;
    for (int c2 = 0; c2 < 16; ++c2) vv += red[t * 17 + c2];
    rsd[t] = rsqrtf(vv * (1.f / 128.f) + EPSF);
  }
  __syncthreads();
  for (int q = 0; q < 8; ++q) {
    int idx = t * 8 + q;
    int m = idx >> 7, cc = idx & 127;
    float val = (A1[m * WP + cc] - mus[m]) * rsd[m] * lng[cc] + lnb[cc];
    out[(size_t)(r0 + m) * 128 + cc] = fmaxf(val, 0.f);
  }
}

// ---------------------------------------------------------------------------
extern "C" void kernel_launch(void* const* d_in, const int* in_sizes, int n_in,
                              void* d_out, int out_size, void* d_ws, size_t ws_size,
                              hipStream_t stream) {
  const float* x     = (const float*)d_in[0];   // [2,512,128]
  const float* adj   = (const float*)d_in[1];   // [2,512,512]
  const float* W_lin = (const float*)d_in[2];   // [128,128]
  const float* b_lin = (const float*)d_in[3];
  const float* W_m1  = (const float*)d_in[4];   // [128,129]
  const float* b_m1  = (const float*)d_in[5];
  const float* W_m2  = (const float*)d_in[6];   // [128,128]
  const float* b_m2  = (const float*)d_in[7];
  const float* W_att = (const float*)d_in[8];   // [1,257]
  const float* b_att = (const float*)d_in[9];
  const float* W_o1  = (const float*)d_in[10];
  const float* b_o1  = (const float*)d_in[11];
  const float* W_o2  = (const float*)d_in[12];
  const float* b_o2  = (const float*)d_in[13];
  const float* ln_g  = (const float*)d_in[14];
  const float* ln_b  = (const float*)d_in[15];

  float* out = (float*)d_out;                   // 131072 out + 524288 adj

  float* ws = (float*)d_ws;
  float* h = ws;                                // BN*128
  float* p = h + BN * 128;                      // BN*128
  float* u = p + BN * 128;                      // BN
  float* v = u + BN;                            // BN
  float* s = v + BN;                            // BN*128
  float* c = s + BN * 128;                      // BN

  const size_t sh_gemm = (size_t)(128 * WP + 16 * WP) * sizeof(float);        // ~74 KB
  const size_t sh_out  = (size_t)(3 * 128 * WP + 2 * 16 * WP + 16 + 16 * 17 + 16 + 16) *
                         sizeof(float);                                        // ~216 KB

  // h = x @ W_lin^T + b_lin
  gemm_wmma<<<BN / 16, 256, sh_gemm, stream>>>(h, x, W_lin, 128, b_lin);
  // p = h @ W_m1[:, :128]^T + b_m1   (pitch 129)
  gemm_wmma<<<BN / 16, 256, sh_gemm, stream>>>(p, h, W_m1, 129, b_m1);
  // per-node attention dots
  uv_kernel<<<BN, 128, 0, stream>>>(h, W_att, u, v);
  // edge aggregation
  pair_agg<<<BN, 256, 0, stream>>>(adj, p, u, v, W_m1, W_att, b_att, s, c);
  // fused output network + residual + LN + relu
  out_net<<<BN / 16, 256, sh_out, stream>>>(out, s, c, h, W_m2, b_m2, W_o1, b_o1,
                                            W_o2, b_o2, ln_g, ln_b);
  // second tuple element: adj passthrough
  hipMemcpyAsync(out + BN * 128, adj, (size_t)BB * NN * NN * sizeof(float),
                 hipMemcpyDeviceToDevice, stream);
}